// GATLayer_49392123904377
// MI455X (gfx1250) — compile-verified
//
#include <hip/hip_runtime.h>
#include <hip/hip_bf16.h>
#include <math.h>

typedef __attribute__((ext_vector_type(2))) float v2f;
typedef __attribute__((ext_vector_type(4))) float v4f;
typedef __attribute__((ext_vector_type(8))) float v8f;

#define BTOT   192              // B*T
#define NN     325              // nodes
#define DD     64               // feature dim
#define HH     4                // heads
#define COLS   256              // H*D
#define MROWS  (NN * BTOT)      // 62400 GEMM rows
#define MTILES (MROWS / 16)     // 3900 (exact)
#define NEG_SLOPE 0.2f

// ---------------------------------------------------------------------------
// Kernel 1: projection GEMM  xp[n*BT+bt, h*64+f] = sum_d x[bt,n,d] * W[d,h,f]
// W resident in LDS (64KB), f32 WMMA 16x16x4, K-loop over D=64.
// ---------------------------------------------------------------------------
__global__ __launch_bounds__(256) void gat_proj_wmma(
    const float* __restrict__ x,    // [BT, N, D]
    const float* __restrict__ W,    // [D, H*D] = [64, 256]
    float* __restrict__ xp)         // [MROWS, COLS]
{
    __shared__ float sW[DD * COLS];        // 65536 bytes
    for (int i = threadIdx.x; i < (DD * COLS) / 4; i += 256) {
        ((v4f*)sW)[i] = ((const v4f*)W)[i];
    }
    __syncthreads();

    const int wave  = threadIdx.x >> 5;
    const int lane  = threadIdx.x & 31;
    const int mtile = blockIdx.x * 8 + wave;
    if (mtile >= MTILES) return;

    const int m0   = mtile * 16;
    const int r    = lane & 15;
    const int kgrp = (lane >> 4) * 2;       // lanes 0-15 -> K{0,1}, 16-31 -> K{2,3}
    const int m    = m0 + r;
    const int n    = m / BTOT;
    const int bt   = m - n * BTOT;
    const float* arow = x + ((size_t)bt * NN + n) * DD;

    // A fragments for all 16 K-steps (each step covers K=4)
    v2f afrag[16];
#pragma unroll
    for (int kk = 0; kk < 16; ++kk)
        afrag[kk] = *(const v2f*)(arow + kk * 4 + kgrp);

    float* orow = xp + (size_t)m0 * COLS;
#pragma unroll 1
    for (int c = 0; c < 16; ++c) {
        v8f acc = {};
#pragma unroll
        for (int kk = 0; kk < 16; ++kk) {
            v2f b;
            b.x = sW[(kk * 4 + kgrp + 0) * COLS + c * 16 + (lane & 15)];
            b.y = sW[(kk * 4 + kgrp + 1) * COLS + c * 16 + (lane & 15)];
            acc = __builtin_amdgcn_wmma_f32_16x16x4_f32(
                false, afrag[kk], false, b, (short)0, acc, false, false);
        }
        // D layout: VGPR v -> row v + 8*(lane>=16), col = lane&15 within tile
        const int col   = c * 16 + (lane & 15);
        const int rbase = (lane >> 4) * 8;
#pragma unroll
        for (int v = 0; v < 8; ++v)
            orow[(size_t)(rbase + v) * COLS + col] = acc[v];
    }
}

// ---------------------------------------------------------------------------
// Kernel 2: attention logits  a_src/a_dst[m, h] = dot(xp[m, h*64:...], att)
// One wave per row m; 8 lanes per head, shfl_xor reduction.
// ---------------------------------------------------------------------------
__global__ __launch_bounds__(256) void gat_att(
    const float* __restrict__ xp,
    const float* __restrict__ att_src,   // [H, D]
    const float* __restrict__ att_dst,   // [H, D]
    float* __restrict__ asrc,            // [MROWS, H]
    float* __restrict__ adst)            // [MROWS, H]
{
    const int gid = blockIdx.x * 256 + threadIdx.x;
    const int m   = gid >> 5;
    if (m >= MROWS) return;
    const int lane = threadIdx.x & 31;
    const int h    = lane >> 3;
    const int sub  = lane & 7;

    const float* p  = xp + (size_t)m * COLS + h * DD + sub * 8;
    const float* as = att_src + h * DD + sub * 8;
    const float* ad = att_dst + h * DD + sub * 8;
    float ssum = 0.f, dsum = 0.f;
#pragma unroll
    for (int i = 0; i < 8; ++i) { float v = p[i]; ssum += v * as[i]; dsum += v * ad[i]; }
#pragma unroll
    for (int o = 4; o >= 1; o >>= 1) {
        ssum += __shfl_xor(ssum, o, 32);
        dsum += __shfl_xor(dsum, o, 32);
    }
    if (sub == 0) {
        asrc[(size_t)m * HH + h] = ssum;
        adst[(size_t)m * HH + h] = dsum;
    }
}

// ---------------------------------------------------------------------------
// CSR build by destination node (self-loops included via deg init = 1).
// ---------------------------------------------------------------------------
__global__ void csr_init(int* __restrict__ deg) {
    int n = blockIdx.x * blockDim.x + threadIdx.x;
    if (n < NN) deg[n] = 1;                        // self loop
}

__global__ void csr_count(const int* __restrict__ ei, int* __restrict__ deg, int E) {
    int e = blockIdx.x * blockDim.x + threadIdx.x;
    if (e < E) atomicAdd(&deg[ei[E + e]], 1);      // row 1 = dst
}

__global__ void csr_scan(const int* __restrict__ deg, int* __restrict__ off,
                         int* __restrict__ cur) {
    if (blockIdx.x == 0 && threadIdx.x == 0) {
        int acc = 0;
        for (int n = 0; n < NN; ++n) { off[n] = acc; cur[n] = acc; acc += deg[n]; }
        off[NN] = acc;
    }
}

__global__ void csr_fill(const int* __restrict__ ei, int* __restrict__ cur,
                         int* __restrict__ csrc, int E) {
    int t = blockIdx.x * blockDim.x + threadIdx.x;
    if (t < E) {
        int s = ei[t], d = ei[E + t];
        csrc[atomicAdd(&cur[d], 1)] = s;
    } else if (t < E + NN) {
        int n = t - E;
        csrc[atomicAdd(&cur[n], 1)] = n;           // self loop src = n
    }
}

// ---------------------------------------------------------------------------
// Kernel 3: per-(node, bt) online-softmax gather aggregation, fused
// head-mean + bias + residual + LayerNorm epilogue.
// Block = 128 threads = 4 waves = 4 heads; lane owns 2 feature dims.
// ---------------------------------------------------------------------------
__global__ __launch_bounds__(128) void gat_agg_ln(
    const float* __restrict__ x,       // [BT, N, D]
    const float* __restrict__ xp,      // [MROWS, COLS]
    const float* __restrict__ asrc,    // [MROWS, H]
    const float* __restrict__ adst,    // [MROWS, H]
    const int* __restrict__ off,
    const int* __restrict__ csrc,
    const float* __restrict__ bias,
    const float* __restrict__ gamma,
    const float* __restrict__ beta,
    float* __restrict__ out)           // [BT, N, D]
{
    __shared__ float heads[HH * DD];
    const int id   = blockIdx.x;       // n*BT + bt
    const int n    = id / BTOT;
    const int bt   = id - n * BTOT;
    const int h    = threadIdx.x >> 5;
    const int lane = threadIdx.x & 31;
    const int d0   = lane * 2;

    const float ad = adst[((size_t)n * BTOT + bt) * HH + h];
    const int kb = off[n], ke = off[n + 1];

    float mrun = -3.402823466e+38f;    // -FLT_MAX
    float srun = 0.f, acc0 = 0.f, acc1 = 0.f;
    for (int k = kb; k < ke; ++k) {
        const int j = csrc[k];
        float a = asrc[((size_t)j * BTOT + bt) * HH + h] + ad;
        a = (a >= 0.f) ? a : NEG_SLOPE * a;
        const float mnew = fmaxf(mrun, a);
        const float corr = __expf(mrun - mnew);
        const float w    = __expf(a - mnew);
        srun = srun * corr + w;
        const float* xj = xp + ((size_t)j * BTOT + bt) * COLS + h * DD + d0;
        acc0 = acc0 * corr + w * xj[0];
        acc1 = acc1 * corr + w * xj[1];
        mrun = mnew;
    }
    const float inv = 1.f / (srun + 1e-16f);
    heads[h * DD + d0]     = acc0 * inv;
    heads[h * DD + d0 + 1] = acc1 * inv;
    __syncthreads();

    if (h == 0) {
        float y0 = 0.f, y1 = 0.f;
#pragma unroll
        for (int hh = 0; hh < HH; ++hh) {
            y0 += heads[hh * DD + d0];
            y1 += heads[hh * DD + d0 + 1];
        }
        const float* xr = x + ((size_t)bt * NN + n) * DD;
        y0 = y0 * 0.25f + bias[d0]     + xr[d0];
        y1 = y1 * 0.25f + bias[d0 + 1] + xr[d0 + 1];
        float s  = y0 + y1;
        float s2 = y0 * y0 + y1 * y1;
#pragma unroll
        for (int o = 16; o >= 1; o >>= 1) {
            s  += __shfl_xor(s,  o, 32);
            s2 += __shfl_xor(s2, o, 32);
        }
        const float mu  = s * (1.f / 64.f);
        const float var = s2 * (1.f / 64.f) - mu * mu;
        const float rs  = rsqrtf(var + 1e-5f);
        float* orow = out + ((size_t)bt * NN + n) * DD;
        orow[d0]     = (y0 - mu) * rs * gamma[d0]     + beta[d0];
        orow[d0 + 1] = (y1 - mu) * rs * gamma[d0 + 1] + beta[d0 + 1];
    }
}

// ---------------------------------------------------------------------------
extern "C" void kernel_launch(void* const* d_in, const int* in_sizes, int n_in,
                              void* d_out, int out_size, void* d_ws, size_t ws_size,
                              hipStream_t stream) {
    const float* x        = (const float*)d_in[0];
    const float* W        = (const float*)d_in[1];
    const float* att_src  = (const float*)d_in[2];
    const float* att_dst  = (const float*)d_in[3];
    const float* bias     = (const float*)d_in[4];
    const float* gamma    = (const float*)d_in[5];
    const float* beta     = (const float*)d_in[6];
    const int*   ei       = (const int*)d_in[7];       // [2, E] int32
    float* out            = (float*)d_out;
    const int E = in_sizes[7] / 2;

    // workspace layout
    char* ws = (char*)d_ws;
    float* xp   = (float*)ws;  ws += sizeof(float) * (size_t)MROWS * COLS;   // 63.9 MB
    float* asrc = (float*)ws;  ws += sizeof(float) * (size_t)MROWS * HH;
    float* adst = (float*)ws;  ws += sizeof(float) * (size_t)MROWS * HH;
    int* deg    = (int*)ws;    ws += sizeof(int) * NN;
    int* off    = (int*)ws;    ws += sizeof(int) * (NN + 1);
    int* cur    = (int*)ws;    ws += sizeof(int) * NN;
    int* csrc   = (int*)ws;    ws += sizeof(int) * (E + NN);

    // 1) projection GEMM (WMMA f32 16x16x4)
    gat_proj_wmma<<<(MTILES + 7) / 8, 256, 0, stream>>>(x, W, xp);

    // 2) attention logits
    gat_att<<<(MROWS * 32 + 255) / 256, 256, 0, stream>>>(xp, att_src, att_dst, asrc, adst);

    // 3) CSR by destination
    csr_init <<<(NN + 255) / 256, 256, 0, stream>>>(deg);
    csr_count<<<(E + 255) / 256, 256, 0, stream>>>(ei, deg, E);
    csr_scan <<<1, 32, 0, stream>>>(deg, off, cur);
    csr_fill <<<(E + NN + 255) / 256, 256, 0, stream>>>(ei, cur, csrc, E);

    // 4) gather aggregation + fused LayerNorm
    gat_agg_ln<<<MROWS, 128, 0, stream>>>(x, xp, asrc, adst, off, csrc,
                                          bias, gamma, beta, out);
}